// GraphNet_37520834297930
// MI455X (gfx1250) — compile-verified
//
#include <hip/hip_runtime.h>

typedef float v2f __attribute__((ext_vector_type(2)));
typedef float v8f __attribute__((ext_vector_type(8)));

constexpr int NN = 50000;   // nodes
constexpr int NE = 400000;  // edges

// Workspace layout (in floats)
constexpr size_t P_OFF   = 0;                      // 50000*384 = 19,200,000
constexpr size_t Q_OFF   = 19200000;               // 150000
constexpr size_t XR_OFF  = 19350000;               // 150000
constexpr size_t SUM_OFF = 19500000;               // 150000
constexpr size_t CNT_OFF = 19650000;               // 50000
constexpr size_t LG_OFF  = 19700000;               // 150000
constexpr size_t ST_OFF  = 19850000;               // 6

__device__ __forceinline__ v8f wmma4(v2f a, v2f b, v8f c) {
  // D = A(16x4,f32) * B(4x16,f32) + C(16x16,f32)
  return __builtin_amdgcn_wmma_f32_16x16x4_f32(false, a, false, b, (short)0, c,
                                               false, false);
}

__device__ __forceinline__ v2f ld2(const float* p) {
  float2 t = *(const float2*)p;
  v2f r;
  r.x = t.x;
  r.y = t.y;
  return r;
}

__device__ __forceinline__ float elu(float x) {
  return x > 0.0f ? x : (__expf(x) - 1.0f);
}

// ---------------------------------------------------------------------------
// P[n, k*3+o] = sum_i x[n,i] * W3[k, i*3+o]   (x:[NN,256], W3:[128,768])
// Block: 64 nodes. 8 waves; each wave: all 4 row-tiles x 3 col-tiles (48 cols).
// ---------------------------------------------------------------------------
__global__ __launch_bounds__(256) void p_kernel(const float* __restrict__ x,
                                                const float* __restrict__ W3,
                                                float* __restrict__ P) {
  const int tid = threadIdx.x;
  const int lane = tid & 31, wid = tid >> 5;
  const int l15 = lane & 15, h16 = lane >> 4;
  const int nb = blockIdx.x * 64;
  const int n0 = wid * 48;

  v8f acc[12] = {};  // [rt*3 + j]

  size_t aoff[4];
#pragma unroll
  for (int rt = 0; rt < 4; ++rt) {
    int mrow = nb + rt * 16 + l15;
    if (mrow > NN - 1) mrow = NN - 1;
    aoff[rt] = (size_t)mrow * 256 + 2 * h16;
  }
  int bbase[3];
#pragma unroll
  for (int j = 0; j < 3; ++j) {
    int n = n0 + j * 16 + l15;
    int kq = n / 3;
    int oq = n - 3 * kq;
    bbase[j] = kq * 768 + oq + 3 * (2 * h16);
  }

  auto loadA = [&](v2f a[4], int k) {
#pragma unroll
    for (int rt = 0; rt < 4; ++rt) a[rt] = ld2(x + aoff[rt] + k);
  };
  auto loadB = [&](v2f b[3], int k) {
#pragma unroll
    for (int j = 0; j < 3; ++j) {
      b[j].x = W3[bbase[j] + k * 3];
      b[j].y = W3[bbase[j] + k * 3 + 3];
    }
  };
  auto mm = [&](v2f a[4], v2f b[3]) {
#pragma unroll
    for (int rt = 0; rt < 4; ++rt)
#pragma unroll
      for (int j = 0; j < 3; ++j)
        acc[rt * 3 + j] = wmma4(a[rt], b[j], acc[rt * 3 + j]);
  };

  v2f a0[4], b0[3], a1[4], b1v[3];
  loadA(a0, 0);
  loadB(b0, 0);
#pragma unroll 1
  for (int k0 = 0; k0 < 248; k0 += 8) {
    loadA(a1, k0 + 4);
    loadB(b1v, k0 + 4);
    mm(a0, b0);
    loadA(a0, k0 + 8);
    loadB(b0, k0 + 8);
    mm(a1, b1v);
  }
  loadA(a1, 252);
  loadB(b1v, 252);
  mm(a0, b0);
  mm(a1, b1v);

  // store (guarded)
#pragma unroll
  for (int rt = 0; rt < 4; ++rt) {
    const int mb = nb + rt * 16 + 8 * h16;
#pragma unroll
    for (int j = 0; j < 3; ++j) {
      int n = n0 + j * 16 + l15;
      float* drow = P + (size_t)mb * 384 + n;
#pragma unroll
      for (int v = 0; v < 8; ++v) {
        if (mb + v < NN) drow[(size_t)v * 384] = acc[rt * 3 + j][v];
      }
    }
  }
}

// ---------------------------------------------------------------------------
// q[n,o] = x[n]·b3[:,o]; xr[n,o] = x[n]·root[:,o] + bias[o]. One wave/node.
// ---------------------------------------------------------------------------
__global__ __launch_bounds__(256) void qxr_kernel(const float* __restrict__ x,
                                                  const float* __restrict__ b3,
                                                  const float* __restrict__ root,
                                                  const float* __restrict__ bias,
                                                  float* __restrict__ q,
                                                  float* __restrict__ xr) {
  const int gw = (blockIdx.x * 256 + threadIdx.x) >> 5;
  const int lane = threadIdx.x & 31;
  if (gw >= NN) return;
  const float* xrow = x + (size_t)gw * 256;
  float q0 = 0, q1 = 0, q2 = 0, r0 = 0, r1 = 0, r2 = 0;
  for (int i = lane; i < 256; i += 32) {
    float xv = xrow[i];
    q0 += xv * b3[3 * i + 0];
    q1 += xv * b3[3 * i + 1];
    q2 += xv * b3[3 * i + 2];
    r0 += xv * root[3 * i + 0];
    r1 += xv * root[3 * i + 1];
    r2 += xv * root[3 * i + 2];
  }
#pragma unroll
  for (int off = 16; off; off >>= 1) {
    q0 += __shfl_xor(q0, off);
    q1 += __shfl_xor(q1, off);
    q2 += __shfl_xor(q2, off);
    r0 += __shfl_xor(r0, off);
    r1 += __shfl_xor(r1, off);
    r2 += __shfl_xor(r2, off);
  }
  if (lane == 0) {
    q[(size_t)gw * 3 + 0] = q0;
    q[(size_t)gw * 3 + 1] = q1;
    q[(size_t)gw * 3 + 2] = q2;
    xr[(size_t)gw * 3 + 0] = r0 + bias[0];
    xr[(size_t)gw * 3 + 1] = r1 + bias[1];
    xr[(size_t)gw * 3 + 2] = r2 + bias[2];
  }
}

// ---------------------------------------------------------------------------
// Fused edge MLP + message + scatter. Block = 64 edges, 256 threads (8 waves).
// LDS: A[64x264] (edge_attr, reused as h2[64x136]) + h1[64x264] = 132 KB.
// ---------------------------------------------------------------------------
__global__ __launch_bounds__(256) void edge_kernel(
    const float* __restrict__ ea, const int* __restrict__ ei,
    const float* __restrict__ W1, const float* __restrict__ b1,
    const float* __restrict__ W2, const float* __restrict__ b2,
    const float* __restrict__ P, const float* __restrict__ q,
    float* __restrict__ sums, float* __restrict__ cnt) {
  extern __shared__ float smem[];
  constexpr int SA = 264, SH = 264, S2 = 136;
  float* As = smem;               // 64*264
  float* Hs = smem + 64 * SA;     // 64*264
  float* H2s = smem;              // overlays As (safe after GEMM1 barrier)

  const int tid = threadIdx.x;
  const int lane = tid & 31, wid = tid >> 5;
  const int l15 = lane & 15, h16 = lane >> 4;
  const int e0 = blockIdx.x * 64;

  // Stage edge_attr tile [64 x 256] with b128 loads
  {
    const float4* s4 = (const float4*)(ea + (size_t)e0 * 256);
#pragma unroll
    for (int i = 0; i < 16; ++i) {
      int lin = tid + i * 256;
      int row = lin >> 4, c4 = lin & 15;
      float4 v = s4[row * 64 + c4];
      float* d = As + row * SA + c4 * 4;
      d[0] = v.x; d[1] = v.y; d[2] = v.z; d[3] = v.w;
    }
  }
  __syncthreads();

  // ---- GEMM1: h1 = ELU(A[64x256] @ W1[256x256] + b1) ----
  // wave covers all 4 row-tiles x 2 col-tiles (cols wid*32 .. wid*32+31)
  {
    const int n0 = wid * 32;
    v8f acc[8] = {};  // [rt*2 + c]
    const float* abase = As + l15 * SA + 2 * h16;
    const int bb = (2 * h16) * 256 + n0 + l15;

    auto loadA = [&](v2f a[4], int k) {
#pragma unroll
      for (int rt = 0; rt < 4; ++rt) a[rt] = ld2(abase + rt * 16 * SA + k);
    };
    auto loadB = [&](v2f b[2], int k) {
#pragma unroll
      for (int c = 0; c < 2; ++c) {
        b[c].x = W1[bb + k * 256 + c * 16];
        b[c].y = W1[bb + k * 256 + c * 16 + 256];
      }
    };
    auto mm = [&](v2f a[4], v2f b[2]) {
#pragma unroll
      for (int rt = 0; rt < 4; ++rt)
#pragma unroll
        for (int c = 0; c < 2; ++c)
          acc[rt * 2 + c] = wmma4(a[rt], b[c], acc[rt * 2 + c]);
    };

    v2f a0[4], b0[2], a1[4], b1v[2];
    loadA(a0, 0);
    loadB(b0, 0);
#pragma unroll 1
    for (int k0 = 0; k0 < 248; k0 += 8) {
      loadA(a1, k0 + 4);
      loadB(b1v, k0 + 4);
      mm(a0, b0);
      loadA(a0, k0 + 8);
      loadB(b0, k0 + 8);
      mm(a1, b1v);
    }
    loadA(a1, 252);
    loadB(b1v, 252);
    mm(a0, b0);
    mm(a1, b1v);

#pragma unroll
    for (int rt = 0; rt < 4; ++rt) {
      const int mb = rt * 16 + 8 * h16;
#pragma unroll
      for (int c = 0; c < 2; ++c) {
        int n = n0 + c * 16 + l15;
        float bias1 = b1[n];
        float* drow = Hs + mb * SH + n;
#pragma unroll
        for (int v = 0; v < 8; ++v) drow[v * SH] = elu(acc[rt * 2 + c][v] + bias1);
      }
    }
  }
  __syncthreads();  // h1 complete; As no longer read anywhere

  // ---- GEMM2: h2 = ELU(h1[64x256] @ W2[256x128] + b2) ----
  // wave covers all 4 row-tiles x 1 col-tile (cols wid*16 .. wid*16+15)
  {
    const int n0 = wid * 16;
    v8f acc[4] = {};
    const float* abase = Hs + l15 * SH + 2 * h16;
    const int bb = (2 * h16) * 128 + n0 + l15;

    auto loadA = [&](v2f a[4], int k) {
#pragma unroll
      for (int rt = 0; rt < 4; ++rt) a[rt] = ld2(abase + rt * 16 * SH + k);
    };
    auto loadB = [&](v2f& b, int k) {
      b.x = W2[bb + k * 128];
      b.y = W2[bb + k * 128 + 128];
    };
    auto mm = [&](v2f a[4], v2f b) {
#pragma unroll
      for (int rt = 0; rt < 4; ++rt) acc[rt] = wmma4(a[rt], b, acc[rt]);
    };

    v2f a0[4], a1[4];
    v2f b0, b1v;
    loadA(a0, 0);
    loadB(b0, 0);
#pragma unroll 1
    for (int k0 = 0; k0 < 248; k0 += 8) {
      loadA(a1, k0 + 4);
      loadB(b1v, k0 + 4);
      mm(a0, b0);
      loadA(a0, k0 + 8);
      loadB(b0, k0 + 8);
      mm(a1, b1v);
    }
    loadA(a1, 252);
    loadB(b1v, 252);
    mm(a0, b0);
    mm(a1, b1v);

    const int n = n0 + l15;
    const float bias2 = b2[n];
#pragma unroll
    for (int rt = 0; rt < 4; ++rt) {
      float* drow = H2s + (rt * 16 + 8 * h16) * S2 + n;
#pragma unroll
      for (int v = 0; v < 8; ++v) drow[v * S2] = elu(acc[rt][v] + bias2);
    }
  }
  __syncthreads();

  // ---- Message: msg[e,o] = h2[e,:]·P[src[e],:,o] + q[src[e],o]; scatter ----
  {
    const int el = tid >> 2, part = tid & 3;  // 4 lanes per edge
    const int e = e0 + el;
    const int sn = ei[e];
    const int dn = ei[NE + e];
    const float* hrow = H2s + el * S2 + part * 32;
    const float* Pp = P + (size_t)sn * 384 + part * 96;  // 32 k's * 3 o's
    float s0 = 0, s1 = 0, s2 = 0;
#pragma unroll 8
    for (int k = 0; k < 32; ++k) {
      float h = hrow[k];
      s0 += h * Pp[3 * k + 0];
      s1 += h * Pp[3 * k + 1];
      s2 += h * Pp[3 * k + 2];
    }
    s0 += __shfl_xor(s0, 1); s0 += __shfl_xor(s0, 2);
    s1 += __shfl_xor(s1, 1); s1 += __shfl_xor(s1, 2);
    s2 += __shfl_xor(s2, 1); s2 += __shfl_xor(s2, 2);
    if (part == 0) {
      atomicAdd(&sums[(size_t)dn * 3 + 0], s0 + q[(size_t)sn * 3 + 0]);
      atomicAdd(&sums[(size_t)dn * 3 + 1], s1 + q[(size_t)sn * 3 + 1]);
      atomicAdd(&sums[(size_t)dn * 3 + 2], s2 + q[(size_t)sn * 3 + 2]);
      atomicAdd(&cnt[dn], 1.0f);
    }
  }
}

// ---------------------------------------------------------------------------
// logits = sums/max(cnt,1) + xr
// ---------------------------------------------------------------------------
__global__ __launch_bounds__(256) void finalize_kernel(
    const float* __restrict__ sums, const float* __restrict__ cnt,
    const float* __restrict__ xr, float* __restrict__ lg) {
  int n = blockIdx.x * 256 + threadIdx.x;
  if (n >= NN) return;
  float inv = 1.0f / fmaxf(cnt[n], 1.0f);
#pragma unroll
  for (int o = 0; o < 3; ++o)
    lg[(size_t)n * 3 + o] = sums[(size_t)n * 3 + o] * inv + xr[(size_t)n * 3 + o];
}

// Column-wise softmax stats over node axis (single block).
__global__ __launch_bounds__(1024) void stats_kernel(const float* __restrict__ lg,
                                                     float* __restrict__ st) {
  __shared__ float sm[1024];
  const int tid = threadIdx.x;
  float m[3] = {-1e30f, -1e30f, -1e30f};
  for (int n = tid; n < NN; n += 1024) {
    m[0] = fmaxf(m[0], lg[(size_t)n * 3 + 0]);
    m[1] = fmaxf(m[1], lg[(size_t)n * 3 + 1]);
    m[2] = fmaxf(m[2], lg[(size_t)n * 3 + 2]);
  }
  float M[3];
#pragma unroll
  for (int o = 0; o < 3; ++o) {
    sm[tid] = m[o];
    __syncthreads();
    for (int s = 512; s; s >>= 1) {
      if (tid < s) sm[tid] = fmaxf(sm[tid], sm[tid + s]);
      __syncthreads();
    }
    M[o] = sm[0];
    __syncthreads();
  }
  float s3[3] = {0.f, 0.f, 0.f};
  for (int n = tid; n < NN; n += 1024) {
    s3[0] += __expf(lg[(size_t)n * 3 + 0] - M[0]);
    s3[1] += __expf(lg[(size_t)n * 3 + 1] - M[1]);
    s3[2] += __expf(lg[(size_t)n * 3 + 2] - M[2]);
  }
#pragma unroll
  for (int o = 0; o < 3; ++o) {
    sm[tid] = s3[o];
    __syncthreads();
    for (int s = 512; s; s >>= 1) {
      if (tid < s) sm[tid] += sm[tid + s];
      __syncthreads();
    }
    if (tid == 0) {
      st[o] = M[o];
      st[3 + o] = sm[0];
    }
    __syncthreads();
  }
}

__global__ __launch_bounds__(256) void apply_kernel(const float* __restrict__ lg,
                                                    const float* __restrict__ st,
                                                    float* __restrict__ out) {
  int idx = blockIdx.x * 256 + threadIdx.x;
  if (idx >= NN * 3) return;
  int o = idx % 3;
  out[idx] = __expf(lg[idx] - st[o]) / st[3 + o];
}

// ---------------------------------------------------------------------------
extern "C" void kernel_launch(void* const* d_in, const int* in_sizes, int n_in,
                              void* d_out, int out_size, void* d_ws,
                              size_t ws_size, hipStream_t stream) {
  (void)in_sizes; (void)n_in; (void)out_size; (void)ws_size;
  const float* x    = (const float*)d_in[0];
  const int*   ei   = (const int*)d_in[1];
  const float* ea   = (const float*)d_in[2];
  const float* W1   = (const float*)d_in[3];
  const float* b1   = (const float*)d_in[4];
  const float* W2   = (const float*)d_in[5];
  const float* b2   = (const float*)d_in[6];
  const float* W3   = (const float*)d_in[7];
  const float* b3   = (const float*)d_in[8];
  const float* root = (const float*)d_in[9];
  const float* bias = (const float*)d_in[10];

  float* ws   = (float*)d_ws;
  float* P    = ws + P_OFF;
  float* q    = ws + Q_OFF;
  float* xr   = ws + XR_OFF;
  float* sums = ws + SUM_OFF;
  float* cnt  = ws + CNT_OFF;
  float* lg   = ws + LG_OFF;
  float* st   = ws + ST_OFF;

  // zero scatter accumulators (sums then cnt are contiguous)
  hipMemsetAsync((void*)(ws + SUM_OFF), 0, (size_t)(150000 + 50000) * sizeof(float),
                 stream);

  p_kernel<<<(NN + 63) / 64, 256, 0, stream>>>(x, W3, P);
  qxr_kernel<<<NN / 8, 256, 0, stream>>>(x, b3, root, bias, q, xr);

  const size_t lds_bytes = (size_t)(64 * 264 * 2) * sizeof(float);  // 132 KB
  edge_kernel<<<NE / 64, 256, lds_bytes, stream>>>(ea, ei, W1, b1, W2, b2, P, q,
                                                   sums, cnt);

  finalize_kernel<<<(NN + 255) / 256, 256, 0, stream>>>(sums, cnt, xr, lg);
  stats_kernel<<<1, 1024, 0, stream>>>(lg, st);
  apply_kernel<<<(NN * 3 + 255) / 256, 256, 0, stream>>>(lg, st, (float*)d_out);
}